// CoupledOscillatorNetwork_19155554140917
// MI455X (gfx1250) — compile-verified
//
#include <hip/hip_runtime.h>

// Coupled oscillator network on 23x23 torus, 2 fields (x,v), B batches.
// One workgroup (128 threads = 4 waves) per batch; state lives in LDS.
// Each inner step's two 3x3 circular convs are computed as 6 accumulating
// 23x23 x 23x23 matmuls (row-shift * column-circulant decomposition) using
// V_WMMA_F32_16X16X4_F32 tiles (full fp32, matching the reference dtype).
// Each wave owns one 16x16 output quadrant -> no cross-wave reduction.
//
// v2: B fragments (circulants), bias fragment, and the wave's own x/v
// quadrant are held in registers across all 320 steps; per-step LDS traffic
// is only the 36 A-fragment loads (shifted state rows written by all waves)
// plus the predicated state write-back.

#define SP   23           // spatial size
#define CH   (2*SP*SP)    // 1058 floats per (batch, time) slice
#define TPB  128          // 4 waves of 32

typedef __attribute__((ext_vector_type(2))) float v2f;
typedef __attribute__((ext_vector_type(8))) float v8f;

__global__ __launch_bounds__(TPB)
void cosc_wmma_kernel(const float* __restrict__ y0,
                      const float* __restrict__ Wc,     // [3][3] coupling
                      const float* __restrict__ bc,     // [1]
                      const float* __restrict__ Wr,     // [3][3] resid
                      const float* __restrict__ br,     // [1]
                      const float* __restrict__ bbar,   // [23][23]
                      const float* __restrict__ dtp,    // [1]
                      const float* __restrict__ alphap, // [1]
                      const float* __restrict__ gammap, // [1]
                      const int*   __restrict__ nstepp, // [1]
                      float* __restrict__ out)
{
    // LDS: state fields (col 23 zero-padded for K padding), circulant B
    // matrices (K rows padded to 24, N cols padded to 32), combined bias.
    // s_cmat/s_bias are only staging: read into registers once, then dead.
    __shared__ float s_state[2][SP][24];   // [field][h][w]
    __shared__ float s_cmat[6][24][32];    // [mat][k][n]
    __shared__ float s_bias[SP][32];       // [h][n]

    const int b   = blockIdx.x;
    const int tid = threadIdx.x;
    const int T   = nstepp[0];

    // Scalar params (identical math in every thread; cached loads).
    const float dt_l = 1.0f / (1.0f + __expf(-dtp[0]));   // sigmoid(dt)
    const float gp   = fmaxf(gammap[0], 0.0f);            // relu(gamma)
    const float ap   = fmaxf(alphap[0], 0.0f);            // relu(alpha)

    // ---- Build the 6 circulant matrices: mat = field*3 + (dh+1) ----
    // C[k][n] = W[dh+1][dw+1] where dw = (k-n) wrapped into {-1,0,1} mod 23.
    {
        float* cflat = &s_cmat[0][0][0];
        for (int e = tid; e < 6 * 24 * 32; e += TPB) {
            int mat = e / (24 * 32);
            int rem = e % (24 * 32);
            int k = rem / 32, n = rem % 32;
            float val = 0.0f;
            if (k < SP && n < SP) {
                int d = k - n;
                if (d == SP - 1)       d = -1;
                else if (d == -(SP-1)) d = 1;
                if (d >= -1 && d <= 1) {
                    const float* Wp = (mat < 3) ? Wc : Wr;
                    val = Wp[(mat % 3) * 3 + (d + 1)];
                }
            }
            cflat[e] = val;
        }
    }
    // ---- Combined bias: b_coupling + b_resid + b_bar (cols >=23 zero) ----
    {
        float* bflat = &s_bias[0][0];
        const float bsum = bc[0] + br[0];
        for (int e = tid; e < SP * 32; e += TPB) {
            int h = e / 32, n = e % 32;
            bflat[e] = (n < SP) ? (bsum + bbar[h * SP + n]) : 0.0f;
        }
    }
    // ---- Initial state from y0, and t=0 trajectory slice ----
    const long long outBase = (long long)b * (long long)(T + 1) * CH;
    for (int e = tid; e < CH; e += TPB) {
        float val = y0[(long long)b * CH + e];
        out[outBase + e] = val;                 // prepend y0 (t = 0)
        int c = e & 1, p = e >> 1;
        s_state[c][p / SP][p % SP] = val;
    }
    if (tid < SP) { s_state[0][tid][SP] = 0.0f; s_state[1][tid][SP] = 0.0f; }
    __syncthreads();

    // ---- Per-lane fragment geometry (wave32) ----
    const int lane  = tid & 31;
    const int wid   = tid >> 5;
    const int mt    = wid >> 1;          // output row tile  (0/1)
    const int nt    = wid & 1;           // output col tile  (0/1)
    const int mrow  = lane & 15;         // A fragment M
    const int hi    = lane >> 4;         // lane half
    const int khalf = hi * 2;            // A/B fragment K base {0,2}
    const int ncol  = nt * 16 + (lane & 15);           // D/B column
    const int ncolC = (ncol < SP) ? ncol : SP;         // clamped state col

    int rIdx[8]; bool rOk[8];
    #pragma unroll
    for (int j = 0; j < 8; ++j) {
        int r = mt * 16 + hi * 8 + j;    // D row for VGPR j (C/D layout)
        rOk[j]  = (r < SP);
        rIdx[j] = rOk[j] ? r : (r - SP); // wrapped (discarded rows) stay in-bounds
    }
    const bool nOk = (ncol < SP);

    // ---- Hoist loop-invariant fragments into registers ----
    // B fragments: 6 matrices x 6 K-chunks, 2 dwords/lane each (72 VGPRs).
    v2f bfrag[36];
    #pragma unroll
    for (int mat = 0; mat < 6; ++mat) {
        const float* cm = &s_cmat[mat][0][0];
        #pragma unroll
        for (int kc = 0; kc < 6; ++kc) {
            const int kk = kc * 4 + khalf;
            bfrag[mat * 6 + kc].x = cm[kk * 32 + ncol];
            bfrag[mat * 6 + kc].y = cm[(kk + 1) * 32 + ncol];
        }
    }
    // Bias fragment in C/D layout (rows >= 23 carry garbage; discarded).
    v8f biasFrag;
    #pragma unroll
    for (int j = 0; j < 8; ++j) biasFrag[j] = s_bias[rIdx[j]][ncol];

    // The wave's own x/v quadrant lives in registers across all steps.
    float xq[8], vq[8];
    #pragma unroll
    for (int j = 0; j < 8; ++j) {
        xq[j] = s_state[0][rIdx[j]][ncolC];
        vq[j] = s_state[1][rIdx[j]][ncolC];
    }

    // Shifted A-row index per dh (shared by both fields).
    int rA[3];
    #pragma unroll
    for (int dhi = 0; dhi < 3; ++dhi) {
        int r = mt * 16 + mrow + (dhi - 1);
        if (r < 0)   r += SP;
        if (r >= SP) r -= SP;
        rA[dhi] = r;
    }

    // ---- Time loops: T outer steps x 10 inner steps ----
    for (int t = 1; t <= T; ++t) {
        for (int s = 0; s < 10; ++s) {
            // Accumulator starts at the combined bias.
            v8f acc = biasFrag;

            // f += rowshift(state[field], dh) @ C[field][dh]
            #pragma unroll
            for (int mat = 0; mat < 6; ++mat) {
                const int field = (mat < 3) ? 0 : 1;
                const float* sf = &s_state[field][rA[mat % 3]][0];
                #pragma unroll
                for (int kc = 0; kc < 6; ++kc) {
                    const int kk = kc * 4 + khalf;     // K index (<=23; 23 is zero pad)
                    v2f a;
                    a.x = sf[kk];
                    a.y = sf[kk + 1];
                    acc = __builtin_amdgcn_wmma_f32_16x16x4_f32(
                        /*neg_a=*/false, a, /*neg_b=*/false, bfrag[mat * 6 + kc],
                        /*c_mod=*/(short)0, acc,
                        /*reuse_a=*/false, /*reuse_b=*/false);
                }
            }

            // Oscillator update on this wave's quadrant (register-resident).
            #pragma unroll
            for (int j = 0; j < 8; ++j) {
                float v2 = vq[j] + dt_l * (acc[j] - gp * xq[j] - ap * vq[j]);
                vq[j] = v2;
                xq[j] = xq[j] + dt_l * v2;
            }
            __syncthreads();   // all waves finished reading old state
            if (nOk) {
                #pragma unroll
                for (int j = 0; j < 8; ++j) {
                    if (rOk[j]) {
                        s_state[0][rIdx[j]][ncol] = xq[j];
                        s_state[1][rIdx[j]][ncol] = vq[j];
                    }
                }
            }
            __syncthreads();   // new state visible to all waves
        }
        // Emit trajectory slice t (coalesced; layout (h*23+w)*2+c).
        const long long ob = outBase + (long long)t * CH;
        for (int e = tid; e < CH; e += TPB) {
            int c = e & 1, p = e >> 1;
            out[ob + e] = s_state[c][p / SP][p % SP];
        }
        // Next step's LDS writes are fenced by its mid-step barrier, so the
        // reads above cannot race with them.
    }
}

extern "C" void kernel_launch(void* const* d_in, const int* in_sizes, int n_in,
                              void* d_out, int out_size, void* d_ws, size_t ws_size,
                              hipStream_t stream) {
    const float* y0     = (const float*)d_in[0];
    const float* Wc     = (const float*)d_in[1];
    const float* bc     = (const float*)d_in[2];
    const float* Wr     = (const float*)d_in[3];
    const float* br     = (const float*)d_in[4];
    const float* bbar   = (const float*)d_in[5];
    const float* dtp    = (const float*)d_in[6];
    const float* alphap = (const float*)d_in[7];
    const float* gammap = (const float*)d_in[8];
    const int*   nstep  = (const int*)d_in[9];
    float* out = (float*)d_out;

    const int B = in_sizes[0] / CH;   // 1024
    cosc_wmma_kernel<<<B, TPB, 0, stream>>>(y0, Wc, bc, Wr, br, bbar,
                                            dtp, alphap, gammap, nstep, out);
    (void)d_ws; (void)ws_size; (void)out_size; (void)n_in;
}